// QuantumPerceptron_65412351918179
// MI455X (gfx1250) — compile-verified
//
#include <hip/hip_runtime.h>
#include <math.h>

typedef __attribute__((ext_vector_type(2))) float v2f;
typedef __attribute__((ext_vector_type(4))) float v4f;
typedef __attribute__((ext_vector_type(8))) float v8f;
typedef int b128_t __attribute__((vector_size(16)));   // matches builtin's pointee type

#define NQ   7
#define DIM  128     // 2^7
#define HALF 64      // even columns only
#define NLAY 4
#define PAD  68      // row padding (floats): 272B rows -> 16B aligned, conflict-free banks

// --- gfx1250 async global->LDS staging (ASYNCcnt path), with safe fallback ---
#if defined(__has_builtin)
#  if __has_builtin(__builtin_amdgcn_global_load_async_to_lds_b128)
#    define HAVE_ASYNC_LDS 1
#  else
#    define HAVE_ASYNC_LDS 0
#  endif
#  if __has_builtin(__builtin_amdgcn_s_wait_asynccnt)
#    define HAVE_WAIT_ASYNC 1
#  else
#    define HAVE_WAIT_ASYNC 0
#  endif
#else
#  define HAVE_ASYNC_LDS 0
#  define HAVE_WAIT_ASYNC 0
#endif

#if HAVE_ASYNC_LDS
__device__ __forceinline__ void async_b128(const float* g, float* l) {
    float* gg = const_cast<float*>(g);
    __builtin_amdgcn_global_load_async_to_lds_b128(
        (__attribute__((address_space(1))) b128_t*)gg,
        (__attribute__((address_space(3))) b128_t*)l,
        0, 0);
}
__device__ __forceinline__ void wait_async() {
#if HAVE_WAIT_ASYNC
    __builtin_amdgcn_s_wait_asynccnt(0);
#else
    asm volatile("s_wait_asynccnt 0" ::: "memory");
#endif
}
#endif

// ---------------------------------------------------------------------------
// Kernel 1 (1 workgroup): evolve the 64 even identity columns through the
// 4*2*7 fused per-qubit rotations G = Rz(g)Rx(b)Rz(a) (a 2x2 complex matrix
// acting on row pairs), then reduce to the 64x64 real symmetric M, and fold
// the 5 identical time columns (H = expm(0) = I) into per-output coefficients.
// ws layout (floats): [0..4095] M row-major, [4096..4097] coef, [4098..4099] bias
// ---------------------------------------------------------------------------
__global__ __launch_bounds__(256)
void qp_build_M(const float* __restrict__ params,
                const float* __restrict__ weight,
                const float* __restrict__ bias,
                float* __restrict__ ws) {
    __shared__ float Er[DIM][HALF];   // 32 KB
    __shared__ float Ei[DIM][HALF];   // 32 KB
    const int tid = threadIdx.x;

    // E = I[:, even columns]
    for (int idx = tid; idx < DIM * HALF; idx += 256) {
        int d = idx >> 6, c = idx & 63;
        Er[d][c] = (d == 2 * c) ? 1.0f : 0.0f;
        Ei[d][c] = 0.0f;
    }
    __syncthreads();

    for (int l = 0; l < NLAY; ++l) {
        for (int sub = 0; sub < 2; ++sub) {
            for (int i = 0; i < NQ; ++i) {
                const int off = l * 14 + sub * 7 + i;
                const float a = params[off];
                const float b = params[56 + off];
                const float g = params[112 + off];
                const float cb = cosf(b),      sb  = sinf(b);
                const float cga = cosf(g + a), sga = sinf(g + a);
                const float cgm = cosf(g - a), sgm = sinf(g - a);
                // G = diag(e^{-ig},e^{ig}) * [[cb,-i sb],[-i sb,cb]] * diag(e^{-ia},e^{ia})
                const float G00r =  cga * cb, G00i = -sga * cb;
                const float G01r = -sgm * sb, G01i = -cgm * sb;
                const float G10r =  sgm * sb, G10i = -cgm * sb;
                const float G11r =  cga * cb, G11i =  sga * cb;
                const int bitpos = 6 - i;          // qubit i -> bit (6-i)
                const int mask   = 1 << bitpos;
                for (int p = tid; p < 64 * 64; p += 256) {
                    const int pr = p >> 6, c = p & 63;           // pair id, column
                    const int low = pr & (mask - 1);
                    const int d0  = ((pr >> bitpos) << (bitpos + 1)) | low;
                    const int d1  = d0 | mask;
                    const float v0r = Er[d0][c], v0i = Ei[d0][c];
                    const float v1r = Er[d1][c], v1i = Ei[d1][c];
                    Er[d0][c] = G00r*v0r - G00i*v0i + G01r*v1r - G01i*v1i;
                    Ei[d0][c] = G00r*v0i + G00i*v0r + G01r*v1i + G01i*v1r;
                    Er[d1][c] = G10r*v0r - G10i*v0i + G11r*v1r - G11i*v1i;
                    Ei[d1][c] = G10r*v0i + G10i*v0r + G11r*v1i + G11i*v1r;
                }
                __syncthreads();
            }
        }
    }

    // M[j][k] = sum_d sign(d) * (Er[d][j]Er[d][k] + Ei[d][j]Ei[d][k])
    for (int idx = tid; idx < 64 * 64; idx += 256) {
        const int j = idx >> 6, k = idx & 63;
        float acc = 0.0f;
        #pragma unroll 4
        for (int d = 0; d < DIM; ++d) {
            const float s = (d & 1) ? -1.0f : 1.0f;
            acc += s * (Er[d][j] * Er[d][k] + Ei[d][j] * Ei[d][k]);
        }
        ws[idx] = acc;
    }
    // coef[o] = sum_j weight[o][j] (all 5 time points identical since H=I)
    if (tid < 2) {
        float c = 0.0f;
        for (int j = 0; j < 5; ++j) c += weight[tid * 5 + j];
        ws[4096 + tid] = c;
        ws[4098 + tid] = bias[tid];
    }
}

// ---------------------------------------------------------------------------
// Kernel 2: per 16-row batch tile (one wave each): T = S_tile (16x64) @ M
// via v_wmma_f32_16x16x4_f32 (4 independent accumulator chains over 16
// K-steps), then e[b] = dot(S[b], T[b]) via 16-lane xor-shuffle reduction,
// finally out[b][o] = tanh(e[b]*coef[o] + bias[o]).
// ---------------------------------------------------------------------------
__global__ __launch_bounds__(256)
void qp_qform(const float* __restrict__ state,
              const float* __restrict__ ws,
              float* __restrict__ out,
              int B) {
    __shared__ __align__(16) float Ml[64][PAD];
    __shared__ __align__(16) float Sl[8][16][PAD];
    __shared__ float el[8][16];

    const int tid  = threadIdx.x;
    const int wave = tid >> 5;
    const int lane = tid & 31;
    const int mlan = lane & 15;         // M/N index of this lane's fragment
    const int kh   = lane >> 4;         // K-half (0 -> K 0..1, 1 -> K 2..3)

    const int rowBase = blockIdx.x * 128 + wave * 16;
    const float* sp = state + (size_t)rowBase * 64;

#if HAVE_ASYNC_LDS
    // M: 1024 float4s across 256 threads -> 4 async B128 per thread
    #pragma unroll
    for (int t = 0; t < 4; ++t) {
        const int f4  = tid + t * 256;       // 0..1023
        const int off = f4 * 4;              // 0..4095
        async_b128(ws + off, &Ml[off >> 6][off & 63]);
    }
    // S tile: 256 float4s per wave -> 8 async B128 per lane
    #pragma unroll
    for (int t = 0; t < 8; ++t) {
        const int off = (lane + t * 32) * 4; // 0..1023
        async_b128(sp + off, &Sl[wave][off >> 6][off & 63]);
    }
    const float c0 = ws[4096], c1 = ws[4097];
    const float b0 = ws[4098], b1 = ws[4099];
    wait_async();
#else
    for (int idx = tid; idx < 64 * 64; idx += 256)
        Ml[idx >> 6][idx & 63] = ws[idx];
    const float c0 = ws[4096], c1 = ws[4097];
    const float b0 = ws[4098], b1 = ws[4099];
    #pragma unroll
    for (int t = 0; t < 8; ++t) {
        const int off = (lane + t * 32) * 4;
        const int r = off >> 6, c = off & 63;
        const v4f v = *(const v4f*)(sp + off);
        Sl[wave][r][c]     = v.x;
        Sl[wave][r][c + 1] = v.y;
        Sl[wave][r][c + 2] = v.z;
        Sl[wave][r][c + 3] = v.w;
    }
#endif
    __syncthreads();

    // T = S @ M ; A frag: lane holds S[mlan][kb..kb+1]; B frag: M[kb..kb+1][n]
    v8f acc0 = {}, acc1 = {}, acc2 = {}, acc3 = {};
    #pragma unroll
    for (int kk = 0; kk < 16; ++kk) {
        const int kb = kk * 4 + kh * 2;
        v2f a;
        a.x = Sl[wave][mlan][kb];
        a.y = Sl[wave][mlan][kb + 1];
        v2f bm0, bm1, bm2, bm3;
        bm0.x = Ml[kb][ 0 + mlan]; bm0.y = Ml[kb + 1][ 0 + mlan];
        bm1.x = Ml[kb][16 + mlan]; bm1.y = Ml[kb + 1][16 + mlan];
        bm2.x = Ml[kb][32 + mlan]; bm2.y = Ml[kb + 1][32 + mlan];
        bm3.x = Ml[kb][48 + mlan]; bm3.y = Ml[kb + 1][48 + mlan];
        acc0 = __builtin_amdgcn_wmma_f32_16x16x4_f32(false, a, false, bm0, (short)0, acc0, false, false);
        acc1 = __builtin_amdgcn_wmma_f32_16x16x4_f32(false, a, false, bm1, (short)0, acc1, false, false);
        acc2 = __builtin_amdgcn_wmma_f32_16x16x4_f32(false, a, false, bm2, (short)0, acc2, false, false);
        acc3 = __builtin_amdgcn_wmma_f32_16x16x4_f32(false, a, false, bm3, (short)0, acc3, false, false);
    }

    // e[row] = sum_col S[row][col] * T[row][col]
    // D layout: vgpr v holds T[row = v + 8*kh][col = ntile*16 + mlan]
    float p[8];
    #pragma unroll
    for (int v = 0; v < 8; ++v) {
        const int r = v + kh * 8;
        p[v] = acc0[v] * Sl[wave][r][ 0 + mlan]
             + acc1[v] * Sl[wave][r][16 + mlan]
             + acc2[v] * Sl[wave][r][32 + mlan]
             + acc3[v] * Sl[wave][r][48 + mlan];
    }
    #pragma unroll
    for (int off = 1; off < 16; off <<= 1) {
        #pragma unroll
        for (int v = 0; v < 8; ++v)
            p[v] += __shfl_xor(p[v], off, 32);
    }
    if (mlan == 0) {
        #pragma unroll
        for (int v = 0; v < 8; ++v) el[wave][kh * 8 + v] = p[v];
    }
    __syncthreads();

    // lanes 0..15 -> output column 0 of rows 0..15; lanes 16..31 -> column 1
    const int row = rowBase + mlan;
    if (row < B) {
        const float e = el[wave][mlan];
        const float coef = kh ? c1 : c0;
        const float bia  = kh ? b1 : b0;
        out[(size_t)row * 2 + kh] = tanhf(e * coef + bia);
    }
}

// ---------------------------------------------------------------------------
extern "C" void kernel_launch(void* const* d_in, const int* in_sizes, int n_in,
                              void* d_out, int out_size, void* d_ws, size_t ws_size,
                              hipStream_t stream) {
    const float* state  = (const float*)d_in[0];   // (B, 64) f32
    const float* params = (const float*)d_in[1];   // (3, 4, 2, 7) f32
    const float* weight = (const float*)d_in[2];   // (2, 5) f32
    const float* bias   = (const float*)d_in[3];   // (2,) f32
    float*       out    = (float*)d_out;           // (B, 2) f32
    float*       ws     = (float*)d_ws;

    const int B = in_sizes[0] / HALF;              // 131072

    qp_build_M<<<1, 256, 0, stream>>>(params, weight, bias, ws);

    const int grid = B / 128;                      // 128 batch rows per block (8 waves x 16)
    qp_qform<<<grid, 256, 0, stream>>>(state, ws, out, B);
}